// BipartiteNeuralMessagePassingLayer_66159676227729
// MI455X (gfx1250) — compile-verified
//
#include <hip/hip_runtime.h>

typedef __attribute__((ext_vector_type(2))) float v2f;
typedef __attribute__((ext_vector_type(8))) float v8f;

#define NA   1024
#define NB   1024
#define ECH  16      // edge MLP output channels (EDGE_DIM)
#define EIN  32      // edge_embeds channels (2*EDGE_DIM)
#define NDIM 32      // node dim
#define JT   16      // j's per wmma tile
#define IBLK 128     // i's per block
#define WAVES 8
#define IPW  (IBLK / WAVES)   // 16 i's per wave
#define XSTRIDE 18            // padded LDS row stride (words), even => b64-aligned, conflict-free

// ---------------------------------------------------------------------------
// Kernel 1: a_part[i][n] = nodes_a[i] @ We1[0:32, n]
//           b_part[j][n] = nodes_b[j] @ We1[32:64, n]
// ---------------------------------------------------------------------------
__global__ __launch_bounds__(256) void precompute_parts(
    const float* __restrict__ na, const float* __restrict__ nb,
    const float* __restrict__ We1,
    float* __restrict__ a_part, float* __restrict__ b_part) {
  int t = blockIdx.x * blockDim.x + threadIdx.x;
  if (t >= 2 * NA * ECH) return;
  int half = t / (NA * ECH);
  int r = (t / ECH) % NA;
  int n = t % ECH;
  const float* x = half ? (nb + r * NDIM) : (na + r * NDIM);
  const float* w = We1 + (half ? 32 : 0) * ECH;
  float s = 0.f;
#pragma unroll
  for (int k = 0; k < 32; ++k) s += x[k] * w[k * ECH + n];
  (half ? b_part : a_part)[r * ECH + n] = s;
}

// ---------------------------------------------------------------------------
// Kernel 2: fused edge MLP + message partial sums.
// Grid (NB/JT, NA/IBLK), 256 threads (8 waves). Wave w: j-tile jt, i-range
// [ib*IBLK + w*IPW, +IPW). 16-edge WMMA tiles: M=j-row, N=channel, K=feature.
// ---------------------------------------------------------------------------
__global__ __launch_bounds__(256) void edge_kernel(
    const float* __restrict__ edge,
    const float* __restrict__ We1, const float* __restrict__ be1,
    const float* __restrict__ We2, const float* __restrict__ be2,
    const float* __restrict__ a_part, const float* __restrict__ b_part,
    float* __restrict__ out_edges,
    float* __restrict__ msga_part,   // [NB/JT][NA][ECH]
    float* __restrict__ msgb_part) { // [NA/IBLK][NB][ECH]
  __shared__ float xpose[WAVES][16 * XSTRIDE];
  __shared__ float mb_red[WAVES][16][ECH];

  const int jt = blockIdx.x;
  const int ib = blockIdx.y;
  const int wave = threadIdx.x >> 5;
  const int lane = threadIdx.x & 31;
  const int m = lane & 15;   // tile row (j) for A-frag / column index
  const int hi = lane >> 4;  // half-wave selector (K pairs / M+8 rows)
  const int n = m;           // output channel for B/C/D frags
  const int j0 = jt * JT;
  const int ibase = ib * IBLK + wave * IPW;

  // Loop-invariant B-fragments: layer-1 edge weights We1[64:96,:], layer-2 We2.
  // B layout (4x16 f32): lanes 0-15 hold K={4s,4s+1}, lanes 16-31 K={4s+2,4s+3}.
  v2f w1[8];
#pragma unroll
  for (int s = 0; s < 8; ++s) {
    int k = 64 + 4 * s + 2 * hi;
    w1[s].x = We1[k * ECH + n];
    w1[s].y = We1[(k + 1) * ECH + n];
  }
  v2f w2[4];
#pragma unroll
  for (int s = 0; s < 4; ++s) {
    int k = 4 * s + 2 * hi;
    w2[s].x = We2[k * ECH + n];
    w2[s].y = We2[(k + 1) * ECH + n];
  }
  const float be1n = be1[n];
  const float be2n = be2[n];

  // C-layout bias rows: VGPR r <-> row M = r + 8*hi, col N = n.
  float bp[8];
#pragma unroll
  for (int r = 0; r < 8; ++r) bp[r] = b_part[(j0 + r + 8 * hi) * ECH + n] + be1n;

  v8f mb;
#pragma unroll
  for (int r = 0; r < 8; ++r) mb[r] = 0.f;

  float* xp = &xpose[wave][0];

  for (int ii = 0; ii < IPW; ++ii) {
    const int i = ibase + ii;
    const float ap = a_part[i * ECH + n];
    v8f c1;
#pragma unroll
    for (int r = 0; r < 8; ++r) c1[r] = bp[r] + ap;

    // A-frag source: edge_embeds[i, j0+m, :] (2KB contiguous tile).
    const float* erow = edge + ((size_t)i * NB + j0 + m) * EIN + 2 * hi;
    if (ii + 1 < IPW) __builtin_prefetch(erow + (size_t)NB * EIN, 0, 0);

    // Layer 1: K=32 -> 8 x v_wmma_f32_16x16x4_f32
#pragma unroll
    for (int s = 0; s < 8; ++s) {
      v2f a;
      a.x = erow[4 * s];
      a.y = erow[4 * s + 1];
      c1 = __builtin_amdgcn_wmma_f32_16x16x4_f32(
          false, a, false, w1[s], (short)0, c1, false, false);
    }
#pragma unroll
    for (int r = 0; r < 8; ++r) c1[r] = fmaxf(c1[r], 0.f);

    // Transpose C-layout h into A-layout via padded LDS tile (per-wave, in-order DS).
#pragma unroll
    for (int r = 0; r < 8; ++r) xp[(r + 8 * hi) * XSTRIDE + n] = c1[r];
    asm volatile("s_wait_dscnt 0" ::: "memory");
    __builtin_amdgcn_wave_barrier();

    v8f c2;
#pragma unroll
    for (int r = 0; r < 8; ++r) c2[r] = be2n;

    // Layer 2: K=16 -> 4 WMMAs
#pragma unroll
    for (int s = 0; s < 4; ++s) {
      v2f a;
      a.x = xp[m * XSTRIDE + 4 * s + 2 * hi];
      a.y = xp[m * XSTRIDE + 4 * s + 2 * hi + 1];
      c2 = __builtin_amdgcn_wmma_f32_16x16x4_f32(
          false, a, false, w2[s], (short)0, c2, false, false);
    }
#pragma unroll
    for (int r = 0; r < 8; ++r) c2[r] = fmaxf(c2[r], 0.f);

    // Store new_edges[i, j0+M, n]
    float* orow = out_edges + ((size_t)i * NB + j0) * ECH;
#pragma unroll
    for (int r = 0; r < 8; ++r) orow[(r + 8 * hi) * ECH + n] = c2[r];

    // msg_b partial: register-accumulate over i (rows = j, cols = channel).
    mb += c2;

    // msg_a partial for this i: sum over the 16 j rows per channel.
    float pa = 0.f;
#pragma unroll
    for (int r = 0; r < 8; ++r) pa += c2[r];
    pa += __shfl_xor(pa, 16);
    if (hi == 0) msga_part[((size_t)jt * NA + i) * ECH + n] = pa;
    asm volatile("s_wait_dscnt 0" ::: "memory");  // keep next iter's LDS stores ordered
  }

  // Deterministic cross-wave combine of msg_b partials (waves cover disjoint i).
#pragma unroll
  for (int r = 0; r < 8; ++r) mb_red[wave][r + 8 * hi][n] = mb[r];
  __syncthreads();
  const int jj = threadIdx.x >> 4;  // 0..15
  const int n2 = threadIdx.x & 15;
  float s = 0.f;
#pragma unroll
  for (int w = 0; w < WAVES; ++w) s += mb_red[w][jj][n2];
  msgb_part[((size_t)ib * NB + j0 + jj) * ECH + n2] = s;
}

// ---------------------------------------------------------------------------
// Kernel 3: deterministic reduction of message partials.
// ---------------------------------------------------------------------------
__global__ __launch_bounds__(256) void reduce_msgs(
    const float* __restrict__ msga_part, const float* __restrict__ msgb_part,
    float* __restrict__ msg_a, float* __restrict__ msg_b) {
  int t = blockIdx.x * blockDim.x + threadIdx.x;
  if (t < NA * ECH) {
    float s = 0.f;
#pragma unroll 4
    for (int jt = 0; jt < NB / JT; ++jt) s += msga_part[(size_t)jt * NA * ECH + t];
    msg_a[t] = s;
  } else if (t < 2 * NA * ECH) {
    int u = t - NA * ECH;
    float s = 0.f;
#pragma unroll
    for (int ibk = 0; ibk < NA / IBLK; ++ibk) s += msgb_part[(size_t)ibk * NB * ECH + u];
    msg_b[u] = s;
  }
}

// ---------------------------------------------------------------------------
// Kernel 4: node MLP (48->32 relu, 32->32 relu), one thread per node row.
// Tiny (10 MFLOP) -> scalar VALU is fine.
// ---------------------------------------------------------------------------
__global__ __launch_bounds__(256) void node_kernel(
    const float* __restrict__ nodes_a, const float* __restrict__ nodes_b,
    const float* __restrict__ msg_a, const float* __restrict__ msg_b,
    const float* __restrict__ Wn1, const float* __restrict__ bn1,
    const float* __restrict__ Wn2, const float* __restrict__ bn2,
    float* __restrict__ new_a, float* __restrict__ new_b) {
  int t = blockIdx.x * blockDim.x + threadIdx.x;
  if (t >= 2 * NA) return;
  int half = t >> 10;
  int r = t & (NA - 1);
  const float* node = half ? (nodes_b + r * NDIM) : (nodes_a + r * NDIM);
  const float* msg = half ? (msg_b + r * ECH) : (msg_a + r * ECH);
  float x[NDIM + ECH];
#pragma unroll
  for (int k = 0; k < NDIM; ++k) x[k] = node[k];
#pragma unroll
  for (int k = 0; k < ECH; ++k) x[NDIM + k] = msg[k];
  float h[NDIM];
#pragma unroll
  for (int nn = 0; nn < NDIM; ++nn) {
    float s = bn1[nn];
#pragma unroll
    for (int k = 0; k < NDIM + ECH; ++k) s += x[k] * Wn1[k * NDIM + nn];
    h[nn] = fmaxf(s, 0.f);
  }
  float* o = (half ? new_b : new_a) + r * NDIM;
#pragma unroll
  for (int nn = 0; nn < NDIM; ++nn) {
    float s = bn2[nn];
#pragma unroll
    for (int k = 0; k < NDIM; ++k) s += h[k] * Wn2[k * NDIM + nn];
    o[nn] = fmaxf(s, 0.f);
  }
}

// ---------------------------------------------------------------------------
extern "C" void kernel_launch(void* const* d_in, const int* in_sizes, int n_in,
                              void* d_out, int out_size, void* d_ws, size_t ws_size,
                              hipStream_t stream) {
  const float* edge_embeds = (const float*)d_in[0];  // (1024,1024,32)
  const float* nodes_a     = (const float*)d_in[1];  // (1024,32)
  const float* nodes_b     = (const float*)d_in[2];  // (1024,32)
  const float* We1 = (const float*)d_in[3];          // (96,16)
  const float* be1 = (const float*)d_in[4];          // (16)
  const float* We2 = (const float*)d_in[5];          // (16,16)
  const float* be2 = (const float*)d_in[6];          // (16)
  const float* Wn1 = (const float*)d_in[7];          // (48,32)
  const float* bn1 = (const float*)d_in[8];          // (32)
  const float* Wn2 = (const float*)d_in[9];          // (32,32)
  const float* bn2 = (const float*)d_in[10];         // (32)

  float* new_edges = (float*)d_out;                       // 1024*1024*16
  float* new_a = new_edges + (size_t)NA * NB * ECH;       // 1024*32
  float* new_b = new_a + (size_t)NA * NDIM;               // 1024*32

  // Workspace layout (floats)
  float* ws = (float*)d_ws;
  float* a_part    = ws;                                   // 16384
  float* b_part    = a_part + NA * ECH;                    // 16384
  float* msga_part = b_part + NB * ECH;                    // 64*16384 = 1M
  float* msgb_part = msga_part + (size_t)(NB / JT) * NA * ECH;  // 8*16384
  float* msg_a     = msgb_part + (size_t)(NA / IBLK) * NB * ECH;
  float* msg_b     = msg_a + NA * ECH;

  // 1) node projections of We1 halves
  precompute_parts<<<(2 * NA * ECH + 255) / 256, 256, 0, stream>>>(
      nodes_a, nodes_b, We1, a_part, b_part);

  // 2) fused edge MLP + message partials
  dim3 grid(NB / JT, NA / IBLK);
  edge_kernel<<<grid, 256, 0, stream>>>(
      edge_embeds, We1, be1, We2, be2, a_part, b_part,
      new_edges, msga_part, msgb_part);

  // 3) reduce partials
  reduce_msgs<<<(2 * NA * ECH + 255) / 256, 256, 0, stream>>>(
      msga_part, msgb_part, msg_a, msg_b);

  // 4) node MLPs
  node_kernel<<<(2 * NA + 255) / 256, 256, 0, stream>>>(
      nodes_a, nodes_b, msg_a, msg_b, Wn1, bn1, Wn2, bn2, new_a, new_b);
}